// WaveletConv_75325136437726
// MI455X (gfx1250) — compile-verified
//
#include <hip/hip_runtime.h>

// ---------------------------------------------------------------------------
// WaveletConv for MI455X (gfx1250): 8-level analysis DWT (fixed 10-tap) +
// 8-level per-channel synthesis IDWT on V_WMMA_F32_16X16X4_F32 via Toeplitz
// (im2col) weight matrices. f32 end-to-end (HBM-bound: ~128 MiB mandatory
// traffic, ~3 GFLOP; ~134 MB workspace is L2-resident).
// Round 2: LDS-staged operands (branchless fills, bank-conflict-free padded
// indexing), no exec-masked per-element loads, uniform store fast path.
// ---------------------------------------------------------------------------

typedef __attribute__((ext_vector_type(2))) float v2f;
typedef __attribute__((ext_vector_type(8))) float v8f;

__constant__ float c_dec_lo[10] = {
    0.003335725285001549f, -0.012580751999015526f, -0.006241490213011705f,
    0.07757149384006515f,  -0.03224486958502952f,  -0.24229488706619015f,
    0.13842814590110342f,   0.7243085284385744f,    0.6038292697974729f,
    0.160102397974125f };
__constant__ float c_dec_hi[10] = {
   -0.160102397974125f,     0.6038292697974729f,   -0.7243085284385744f,
    0.13842814590110342f,   0.24229488706619015f,  -0.03224486958502952f,
   -0.07757149384006515f,  -0.006241490213011705f,  0.012580751999015526f,
    0.003335725285001549f };

// ---------------------------------------------------------------------------
// Analysis level: 256 outputs per block; stage 520-float input window in LDS.
// out[m] = sum_k in[2m + k - 5] * w[k]   (lax conv = cross-correlation)
// ---------------------------------------------------------------------------
#define AN_BLK 256
#define AN_WIN 520   // window covers 2*255 + 9 = 519

__global__ __launch_bounds__(256) void dwt_analysis_level(
    const float* __restrict__ in, float* __restrict__ lo_out,
    float* __restrict__ hi_out, int n_in, int n_out,
    int in_stride, int lo_stride, int hi_stride)
{
  __shared__ float sx[AN_WIN];
  const int c   = blockIdx.y;
  const int tid = threadIdx.x;
  const int Mb  = blockIdx.x * AN_BLK;
  const float* __restrict__ x = in + (size_t)c * in_stride;

  const int wb = 2 * Mb - 5;
  if (wb >= 0 && wb + AN_WIN <= n_in) {          // block-uniform fast path
    for (int j = tid; j < AN_WIN; j += AN_BLK) sx[j] = x[wb + j];
  } else {                                        // edge: clamp + mask-mul
    for (int j = tid; j < AN_WIN; j += AN_BLK) {
      const int g  = wb + j;
      const int gc = g < 0 ? 0 : (g > n_in - 1 ? n_in - 1 : g);
      const float msk = ((unsigned)g < (unsigned)n_in) ? 1.0f : 0.0f;
      sx[j] = x[gc] * msk;
    }
  }
  __syncthreads();

  const int m = Mb + tid;
  if (m >= n_out) return;
  const int li = 2 * tid;
  float alo = 0.0f, ahi = 0.0f;
#pragma unroll
  for (int k = 0; k < 10; ++k) {
    const float v = sx[li + k];
    alo = fmaf(v, c_dec_lo[k], alo);
    ahi = fmaf(v, c_dec_hi[k], ahi);
  }
  lo_out[(size_t)c * lo_stride + m] = alo;
  hi_out[(size_t)c * hi_stride + m] = ahi;
}

// ---------------------------------------------------------------------------
// Synthesis level via WMMA. Block = 8 waves x 256 m-positions = 2048 m.
//   out[2m]   = sum_r prev[m+r-7]*wl[2r+1] + hi[m+r-7]*wh[2r+1]
//   out[2m+1] = sum_r prev[m+r-7]*wl[2r]   + hi[m+r-7]*wh[2r]
// Per wave: D(16x16) tiles, A(i,k)=sig[m0+16i+k-7] (16x32), Toeplitz
// W(k,n)=w[2(k-n)(+1)] for 0<=k-n<=15; K=32 as 8 chained K=4 f32 WMMAs.
// Operands staged in LDS with +1-per-16 padding (conflict-free stride-16).
// ---------------------------------------------------------------------------
#define SY_MBLK 2048
#define SY_WIN  2072                       // max read index 2064, rounded up
#define PIDX(j) ((j) + ((j) >> 4))         // bank-conflict padding
#define SY_LDS  (SY_WIN + (SY_WIN >> 4))   // 2201 floats per array

__global__ __launch_bounds__(256) void idwt_synth_level_wmma(
    const float* __restrict__ prev, const float* __restrict__ hi,
    const float* __restrict__ filt, float* __restrict__ out,
    int level, int n, int store_len,
    int prev_stride, int hi_stride, int out_stride)
{
  __shared__ float sp[SY_LDS];
  __shared__ float sh[SY_LDS];
  __shared__ float swl[32];
  __shared__ float swh[32];

  const int c   = blockIdx.y;
  const int tid = threadIdx.x;
  const int M0  = blockIdx.x * SY_MBLK;

  const float* __restrict__ pv  = prev + (size_t)c * prev_stride;
  const float* __restrict__ hv  = hi   + (size_t)c * hi_stride;
  const float* __restrict__ wlp = filt + (((size_t)c * 8 + level) * 2) * 32;

  // Prefetch next block's window into L2 (global_prefetch_b8; OOB dropped).
  __builtin_prefetch(pv + M0 + SY_MBLK, 0, 0);
  __builtin_prefetch(hv + M0 + SY_MBLK, 0, 0);

  // ---- cooperative LDS fill (all waves participate, then barrier) --------
  const int wb = M0 - 7;
  if (wb >= 0 && wb + SY_WIN <= n) {             // block-uniform fast path
    for (int j = tid; j < SY_WIN; j += 256) {
      sp[PIDX(j)] = pv[wb + j];
      sh[PIDX(j)] = hv[wb + j];
    }
  } else {                                        // edge: clamp + mask-mul
    for (int j = tid; j < SY_WIN; j += 256) {
      const int g  = wb + j;
      const int gc = g < 0 ? 0 : (g > n - 1 ? n - 1 : g);
      const float msk = ((unsigned)g < (unsigned)n) ? 1.0f : 0.0f;
      sp[PIDX(j)] = pv[gc] * msk;
      sh[PIDX(j)] = hv[gc] * msk;
    }
  }
  if (tid < 64) {                                 // 64 filter taps -> LDS
    const float fv = wlp[tid];
    if (tid < 32) swl[tid] = fv; else swh[tid - 32] = fv;
  }
  __syncthreads();

  const int lane = tid & 31;
  const int wv   = tid >> 5;
  const int m0   = M0 + wv * 256;
  if (m0 >= n) return;   // wave-uniform: EXEC all-1s for WMMA below

  const int nn    = lane & 15;   // A row M / B,D column N for this lane
  const int h16   = lane >> 4;   // lane half: K sub-pair select / M+8
  const int abase = (m0 - M0) + 16 * nn;

  v8f de   = {};   // even-phase D accumulator
  v8f dodd = {};   // odd-phase D accumulator

#pragma unroll
  for (int kc = 0; kc < 8; ++kc) {
    const int kb = 4 * kc + 2 * h16;   // K index of .x per ISA A layout
    const int li = abase + kb;
    v2f alo, ahi2, ble, blo, bhe, bho;
    alo.x  = sp[PIDX(li)];  alo.y  = sp[PIDX(li + 1)];
    ahi2.x = sh[PIDX(li)];  ahi2.y = sh[PIDX(li + 1)];

    const int d0 = kb - nn, d1 = d0 + 1;          // Toeplitz diagonal
    const bool v0 = (unsigned)d0 < 16u;
    const bool v1 = (unsigned)d1 < 16u;
    const int i0 = v0 ? 2 * d0 : 0;
    const int i1 = v1 ? 2 * d1 : 0;
    const float f0 = v0 ? 1.0f : 0.0f;
    const float f1 = v1 ? 1.0f : 0.0f;
    ble.x = swl[i0 + 1] * f0;  ble.y = swl[i1 + 1] * f1;
    blo.x = swl[i0]     * f0;  blo.y = swl[i1]     * f1;
    bhe.x = swh[i0 + 1] * f0;  bhe.y = swh[i1 + 1] * f1;
    bho.x = swh[i0]     * f0;  bho.y = swh[i1]     * f1;

    de   = __builtin_amdgcn_wmma_f32_16x16x4_f32(false, alo,  false, ble,
                                                 (short)0, de,   false, false);
    de   = __builtin_amdgcn_wmma_f32_16x16x4_f32(false, ahi2, false, bhe,
                                                 (short)0, de,   false, false);
    dodd = __builtin_amdgcn_wmma_f32_16x16x4_f32(false, alo,  false, blo,
                                                 (short)0, dodd, false, false);
    dodd = __builtin_amdgcn_wmma_f32_16x16x4_f32(false, ahi2, false, bho,
                                                 (short)0, dodd, false, false);
  }

  // D layout: VGPR r holds (M=r, N=lane) lanes 0-15, (M=r+8) lanes 16-31.
  float* __restrict__ op = out + (size_t)c * out_stride;
  const bool full = (2 * (m0 + 255) + 1) < store_len;   // wave-uniform
  if (full) {
#pragma unroll
    for (int r = 0; r < 8; ++r) {
      const int m = m0 + 16 * (r + 8 * h16) + nn;
      float2 vv; vv.x = de[r]; vv.y = dodd[r];
      *reinterpret_cast<float2*>(op + 2 * m) = vv;   // coalesced pair store
    }
  } else {
#pragma unroll
    for (int r = 0; r < 8; ++r) {
      const int m = m0 + 16 * (r + 8 * h16) + nn;
      const int t = 2 * m;
      if (t + 1 < store_len) {
        float2 vv; vv.x = de[r]; vv.y = dodd[r];
        *reinterpret_cast<float2*>(op + t) = vv;
      } else if (t < store_len) {
        op[t] = de[r];                              // last even-only sample
      }
    }
  }
}

// ---------------------------------------------------------------------------
extern "C" void kernel_launch(void* const* d_in, const int* in_sizes, int n_in,
                              void* d_out, int out_size, void* d_ws, size_t ws_size,
                              hipStream_t stream) {
  const float* x    = (const float*)d_in[0];
  const float* filt = (const float*)d_in[1];
  float* out = (float*)d_out;
  float* ws  = (float*)d_ws;

  const int B = 128;
  int N[9];
  N[0] = 131072;
  for (int l = 1; l <= 8; ++l) N[l] = N[l - 1] / 2 + 1;  // 65537 ... 513

  const int PP = N[1];  // 65537: max ping-pong length per channel
  size_t off = 0;
  float* pp[2];
  pp[0] = ws + off; off += (size_t)B * PP;
  pp[1] = ws + off; off += (size_t)B * PP;
  float* hib[9];
  for (int l = 1; l <= 8; ++l) { hib[l] = ws + off; off += (size_t)B * N[l]; }
  // total ~134 MB — resident in the 192 MB L2 across launches

  // ---- analysis: 8 levels, lo ping-pongs pp[0]/pp[1], hi kept per level ----
  const float* cur = x;
  int cur_stride = N[0];
  for (int l = 1; l <= 8; ++l) {
    float* lo_dst = pp[(l - 1) & 1];
    dim3 grid((N[l] + AN_BLK - 1) / AN_BLK, B);
    dwt_analysis_level<<<grid, 256, 0, stream>>>(
        cur, lo_dst, hib[l], N[l - 1], N[l], cur_stride, PP, N[l]);
    cur = lo_dst;
    cur_stride = PP;
  }

  // ---- synthesis: prev starts as lo_8 (in pp[1]); WMMA per level ----------
  const float* prev = pp[1];
  int prev_stride = PP;
  for (int i = 7; i >= 0; --i) {
    const int Nn = N[i + 1];   // input length of this level
    float* dst; int out_stride, store_len;
    if (i == 0) { dst = out;              out_stride = N[0]; store_len = 2 * Nn - 2; }
    else        { dst = pp[(7 - i) & 1];  out_stride = PP;   store_len = 2 * Nn - 1; }
    dim3 grid((Nn + SY_MBLK - 1) / SY_MBLK, B);   // 8 waves/block, 256 m/wave
    idwt_synth_level_wmma<<<grid, 256, 0, stream>>>(
        prev, hib[i + 1], filt, dst, i, Nn, store_len,
        prev_stride, N[i + 1], out_stride);
    prev = dst;
    prev_stride = out_stride;
  }

  (void)in_sizes; (void)n_in; (void)out_size; (void)ws_size;
}